// uSNC_unpool_120259084529
// MI455X (gfx1250) — compile-verified
//
#include <hip/hip_runtime.h>

#define HP 176
#define WP 608
#define H2 352
#define W2 1216
#define PH 8
#define PW 32
#define PHH (PH + 2)
#define PWW (PW + 2)
#define SKH (2 * PH + 6)
#define SKW (2 * PW + 6)
#define SPOOL (HP * WP)
#define S2 (H2 * W2)
#define EPSF 1e-20f

// ---- LDS layout (floats) ----
constexpr int OFF_W     = 0;                       // 196 weights (pad to 200)
constexpr int SKPLANE   = SKH * SKW;               // 1540
constexpr int OFF_RSKIP = 200;                     // raw scs_skip halo, 2 planes
constexpr int OFF_WSL   = OFF_RSKIP + 2 * SKPLANE; // w_s interleaved [pix][ch], 3080
constexpr int PPLANE    = PHH * PWW;               // 340
constexpr int OFF_RPOOL = OFF_WSL + 2 * SKPLANE;   // 8 staged pooled planes
constexpr int OFF_P     = OFF_RPOOL + 8 * PPLANE;  // pooled struct [pos][12]
constexpr int SMEM_FLOATS = OFF_P + 12 * PPLANE;   // 13160 floats = 52,640 B

typedef __attribute__((ext_vector_type(4))) unsigned int u32x4_t;
typedef __attribute__((ext_vector_type(8))) int i32x8_t;
typedef __attribute__((ext_vector_type(4))) int i32x4_t;

// -------- tiny weight-prep kernel (runs once per launch, 1 thread) --------
__device__ inline float softplusf(float x) {
  if (x > 20.f) return x;
  return log1pf(expf(x));
}

__global__ void prep_kernel(const float* __restrict__ wch_d, const float* __restrict__ wsp_d,
                            const float* __restrict__ wsk_d, const float* __restrict__ wch_s,
                            const float* __restrict__ wsp_s, const float* __restrict__ wsk_s,
                            float* __restrict__ wsout) {
  if (threadIdx.x != 0 || blockIdx.x != 0) return;
  const int A[2][2] = {{3, 1}, {2, 0}};  // kh index per (output parity, tap)
  float wfull[2][2][2][4][4];            // [conv d/s][p][q][kh][kw], w[p][q]=wc[p][q]*wsp[q]
  const float* wch[2] = {wch_d, wch_s};
  const float* wsp[2] = {wsp_d, wsp_s};
  for (int cv = 0; cv < 2; ++cv) {
    float wc[2][2];
    for (int p = 0; p < 2; ++p) {
      for (int q = 0; q < 2; ++q) wc[p][q] = softplusf(wch[cv][p * 2 + q]);
      float s = wc[p][0] + wc[p][1];
      wc[p][0] /= s; wc[p][1] /= s;
    }
    float wsq[2][4][4];
    for (int q = 0; q < 2; ++q) {
      float s = 0.f;
      for (int kh = 0; kh < 4; ++kh) {
        float c0 = softplusf(wsp[cv][q * 8 + kh * 2 + 0]);
        float c1 = softplusf(wsp[cv][q * 8 + kh * 2 + 1]);
        wsq[q][kh][0] = c0; wsq[q][kh][1] = c1; wsq[q][kh][2] = c1; wsq[q][kh][3] = c0;
        s += 2.f * (c0 + c1);
      }
      for (int kh = 0; kh < 4; ++kh)
        for (int kw = 0; kw < 4; ++kw) wsq[q][kh][kw] /= s;
    }
    for (int p = 0; p < 2; ++p)
      for (int q = 0; q < 2; ++q)
        for (int kh = 0; kh < 4; ++kh)
          for (int kw = 0; kw < 4; ++kw) wfull[cv][p][q][kh][kw] = wc[p][q] * wsq[q][kh][kw];
  }
  // forward-conv table (w_conv_s as OIHW): [(kh*4+kw)*4 + o*2 + i]
  for (int kh = 0; kh < 4; ++kh)
    for (int kw = 0; kw < 4; ++kw)
      for (int o = 0; o < 2; ++o)
        for (int i = 0; i < 2; ++i)
          wsout[(kh * 4 + kw) * 4 + o * 2 + i] = wfull[1][o][i][kh][kw];
  // parity-resolved transpose-conv tables: out[b] += W[a][b][A[py][dh]][A[px][dw]] * v[a]
  for (int cv = 0; cv < 2; ++cv)
    for (int py = 0; py < 2; ++py)
      for (int px = 0; px < 2; ++px)
        for (int dh = 0; dh < 2; ++dh)
          for (int dw = 0; dw < 2; ++dw)
            for (int a = 0; a < 2; ++a)
              for (int b = 0; b < 2; ++b)
                wsout[64 + cv * 64 + (py * 2 + px) * 16 + (dh * 2 + dw) * 4 + a * 2 + b] =
                    wfull[cv][a][b][A[py][dh]][A[px][dw]];
  for (int c = 0; c < 2; ++c) {
    wsout[192 + c] = 1.f / (1.f + expf(-wsk_d[c]));  // sigmoid(w_skip_d)
    wsout[194 + c] = 1.f / (1.f + expf(-wsk_s[c]));  // sigmoid(w_skip_s)
  }
}

// -------- TDM 2-D tile load: one DMA descriptor, global -> LDS --------
__device__ inline void tdm_load_2d(const float* g, float* lds, int TW, int TH, int rowStride) {
#if defined(__HIP_DEVICE_COMPILE__) && __has_builtin(__builtin_amdgcn_tensor_load_to_lds)
  unsigned long long ga = (unsigned long long)(size_t)g;
  unsigned ldsOff = (unsigned)(size_t)lds;  // flat LDS addr: low 32 bits = LDS offset
  u32x4_t g0;
  g0.x = 1u;                                           // count = 1 valid descriptor
  g0.y = ldsOff;                                       // lds_addr
  g0.z = (unsigned)ga;                                 // global_addr[31:0]
  g0.w = (unsigned)((ga >> 32) & 0x1FFFFFFULL) | (2u << 30);  // addr[56:32] | type=2
  i32x8_t g1;
  g1.s0 = (int)(2u << 16);                                       // data_size = 4B
  g1.s1 = (int)(((unsigned)TW) << 16);                           // tensor_dim0 lo
  g1.s2 = (int)((((unsigned)TW) >> 16) | (((unsigned)TH) << 16));// dim0 hi | dim1 lo
  g1.s3 = (int)((((unsigned)TH) >> 16) | (((unsigned)TW) << 16));// dim1 hi | tile_dim0
  g1.s4 = (int)(unsigned)TH;                                     // tile_dim1 (tile_dim2=0 -> 2D)
  g1.s5 = rowStride;                                             // tensor_dim0_stride
  g1.s6 = 0; g1.s7 = 0;
  i32x4_t z4 = {0, 0, 0, 0};
  i32x8_t z8 = {0, 0, 0, 0, 0, 0, 0, 0};
  // 6-arg form (clang-23 / therock-10.0): (g0, g1, g2, g3, extra, cpol)
  __builtin_amdgcn_tensor_load_to_lds(g0, g1, z4, z4, z8, 0);
#else
  // fallback: wave-0 scalar staging (correctness-equivalent)
  int lane = (int)(threadIdx.x & 31u);
  for (int i = lane; i < TW * TH; i += 32) lds[i] = g[(size_t)(i / TW) * rowStride + (i % TW)];
#endif
}

__device__ inline void wait_tdm() {
#if defined(__HIP_DEVICE_COMPILE__) && __has_builtin(__builtin_amdgcn_s_wait_tensorcnt)
  __builtin_amdgcn_s_wait_tensorcnt(0);
#endif
}

// -------- fused unpool kernel --------
__global__ __launch_bounds__(256) void unpool_main(
    const float* __restrict__ x_pool, const float* __restrict__ scs_pool,
    const float* __restrict__ cs_pool, const float* __restrict__ x_skip,
    const float* __restrict__ scs_skip, const float* __restrict__ cs_skip,
    const float* __restrict__ wsbuf, float* __restrict__ out_x,
    float* __restrict__ out_scs, float* __restrict__ out_cs) {
  __shared__ float smem[SMEM_FLOATS];
  const int tid = (int)threadIdx.x;
  const int w0 = (int)blockIdx.x * PW;
  const int h0 = (int)blockIdx.y * PH;
  const int bb = (int)blockIdx.z;

  // clamped staging windows (so TDM never reads out of the tensor)
  const int sys = 2 * h0 - 3, sxs = 2 * w0 - 3;
  const int cys = max(sys, 0), cye = min(2 * h0 + 2 * PH + 3, H2);
  const int cxs = max(sxs, 0), cxe = min(2 * w0 + 2 * PW + 3, W2);
  const int THs = cye - cys, TWs = cxe - cxs;
  const int pys = max(h0 - 1, 0), pye = min(h0 + PH + 1, HP);
  const int pxs = max(w0 - 1, 0), pxe = min(w0 + PW + 1, WP);
  const int THp = pye - pys, TWp = pxe - pxs;

  // stage weight tables into LDS (concurrent with TDM issue)
  if (tid < 196) smem[OFF_W + tid] = wsbuf[tid];

  // prefetch the streamed output-phase operands (cs_skip + x_skip tile rows)
  if (tid < 192) {
    int plane = tid >> 5, rem = tid & 31;
    int row = rem >> 1, half = rem & 1;
    size_t off = (size_t)(2 * h0 + row) * W2 + 2 * w0 + half * 32;
    const float* p;
    if (plane < 2) p = cs_skip + (size_t)(bb * 2 + plane) * S2 + off;
    else {
      int r = (plane - 2) >> 1, ch = (plane - 2) & 1;
      p = x_skip + (size_t)((r * 8 + bb) * 2 + ch) * S2 + off;
    }
    __builtin_prefetch(p, 0, 1);
  }

  // wave 0 issues all async tensor DMA loads, then waits TENSORcnt
  if ((tid >> 5) == 0) {
    for (int ch = 0; ch < 2; ++ch)
      tdm_load_2d(scs_skip + ((size_t)(bb * 2 + ch) * H2 + cys) * W2 + cxs,
                  &smem[OFF_RSKIP + ch * SKPLANE], TWs, THs, W2);
    const float* pbase[8] = {
        scs_pool + (size_t)(bb * 2 + 0) * SPOOL, scs_pool + (size_t)(bb * 2 + 1) * SPOOL,
        cs_pool + (size_t)(bb * 2 + 0) * SPOOL,  cs_pool + (size_t)(bb * 2 + 1) * SPOOL,
        x_pool + (size_t)(bb * 2 + 0) * SPOOL,   x_pool + (size_t)(bb * 2 + 1) * SPOOL,
        x_pool + (size_t)((8 + bb) * 2 + 0) * SPOOL,
        x_pool + (size_t)((8 + bb) * 2 + 1) * SPOOL};
    for (int pl = 0; pl < 8; ++pl)
      tdm_load_2d(pbase[pl] + (size_t)pys * WP + pxs, &smem[OFF_RPOOL + pl * PPLANE], TWp, THp, WP);
    wait_tdm();
  }
  __syncthreads();

  // Phase A: w_s = 1/(scs_skip+eps) with zero padding (conv zero-pads w_s)
  for (int idx = tid; idx < SKH * SKW; idx += 256) {
    int r = idx / SKW, c = idx % SKW;
    int gy = sys + r, gx = sxs + c;
    bool ok = (gy >= 0) & (gy < H2) & (gx >= 0) & (gx < W2);
    int si = ok ? (gy - cys) * TWs + (gx - cxs) : 0;
    float v0 = smem[OFF_RSKIP + si];
    float v1 = smem[OFF_RSKIP + SKPLANE + si];
    smem[OFF_WSL + idx * 2 + 0] = ok ? 1.f / (v0 + EPSF) : 0.f;
    smem[OFF_WSL + idx * 2 + 1] = ok ? 1.f / (v1 + EPSF) : 0.f;
  }
  // Phase A: pooled struct {scs,cs, x*scs (both batch copies)} with zero padding
  for (int idx = tid; idx < PPLANE; idx += 256) {
    int ty = idx / PWW, tx = idx % PWW;
    int gh = h0 - 1 + ty, gw = w0 - 1 + tx;
    bool ok = (gh >= 0) & (gh < HP) & (gw >= 0) & (gw < WP);
    int si = ok ? (gh - pys) * TWp + (gw - pxs) : 0;
    float m = ok ? 1.f : 0.f;
    float scs0 = m * smem[OFF_RPOOL + 0 * PPLANE + si];
    float scs1 = m * smem[OFF_RPOOL + 1 * PPLANE + si];
    float cs0 = m * smem[OFF_RPOOL + 2 * PPLANE + si];
    float cs1 = m * smem[OFF_RPOOL + 3 * PPLANE + si];
    float x00 = m * smem[OFF_RPOOL + 4 * PPLANE + si];
    float x01 = m * smem[OFF_RPOOL + 5 * PPLANE + si];
    float x10 = m * smem[OFF_RPOOL + 6 * PPLANE + si];
    float x11 = m * smem[OFF_RPOOL + 7 * PPLANE + si];
    float* P = &smem[OFF_P + idx * 12];
    P[0] = scs0; P[1] = scs1; P[2] = cs0; P[3] = cs1;
    P[4] = x00 * scs0; P[5] = x01 * scs1; P[6] = x10 * scs0; P[7] = x11 * scs1;
  }
  __syncthreads();

  // Phase B: w_s_sum (stride-2 fwd conv of w_s) and q = {cs,scs}/w_s_sum
  for (int idx = tid; idx < PPLANE; idx += 256) {
    int ty = idx / PWW, tx = idx % PWW;
    float ws0 = EPSF, ws1 = EPSF;
#pragma unroll
    for (int kh = 0; kh < 4; ++kh)
#pragma unroll
      for (int kw = 0; kw < 4; ++kw) {
        const float4 w = *(const float4*)&smem[OFF_W + (kh * 4 + kw) * 4];
        int si = ((2 * ty + kh) * SKW + (2 * tx + kw)) * 2;
        float a0 = smem[OFF_WSL + si], a1 = smem[OFF_WSL + si + 1];
        ws0 += w.x * a0 + w.y * a1;
        ws1 += w.z * a0 + w.w * a1;
      }
    float r0 = 1.f / ws0, r1 = 1.f / ws1;
    float* P = &smem[OFF_P + idx * 12];
    P[8] = P[2] * r0; P[9] = P[3] * r1;    // q_cs
    P[10] = P[0] * r0; P[11] = P[1] * r1;  // q_scs
  }
  __syncthreads();

  // Output phase: one output pixel per (thread, iter); 2x2 taps per pixel
  for (int p = tid; p < 4 * PH * PW; p += 256) {
    int lx = p & (2 * PW - 1);
    int ly = p >> 6;
    int y = 2 * h0 + ly, x = 2 * w0 + lx;
    int py = ly & 1, px = lx & 1;
    int hb = ly >> 1, wb = lx >> 1;
    float acc[7][2] = {};  // scs_d, xn0, xn1, cs_unf, scs_unf, cs_fn, scs_fn
    const float* WD = &smem[OFF_W + 64 + (py * 2 + px) * 16];
    const float* WS = &smem[OFF_W + 128 + (py * 2 + px) * 16];
#pragma unroll
    for (int dh = 0; dh < 2; ++dh)
#pragma unroll
      for (int dw = 0; dw < 2; ++dw) {
        int pos = (hb + py + dh) * PWW + (wb + px + dw);
        const float4 v0 = *(const float4*)&smem[OFF_P + pos * 12];
        const float4 v1 = *(const float4*)&smem[OFF_P + pos * 12 + 4];
        const float4 v2 = *(const float4*)&smem[OFF_P + pos * 12 + 8];
        const float4 wd = *(const float4*)&WD[(dh * 2 + dw) * 4];
        const float4 wv = *(const float4*)&WS[(dh * 2 + dw) * 4];
        acc[0][0] += wd.x * v0.x + wd.z * v0.y; acc[0][1] += wd.y * v0.x + wd.w * v0.y;
        acc[1][0] += wd.x * v1.x + wd.z * v1.y; acc[1][1] += wd.y * v1.x + wd.w * v1.y;
        acc[2][0] += wd.x * v1.z + wd.z * v1.w; acc[2][1] += wd.y * v1.z + wd.w * v1.w;
        acc[3][0] += wv.x * v0.z + wv.z * v0.w; acc[3][1] += wv.y * v0.z + wv.w * v0.w;
        acc[4][0] += wv.x * v0.x + wv.z * v0.y; acc[4][1] += wv.y * v0.x + wv.w * v0.y;
        acc[5][0] += wv.x * v2.x + wv.z * v2.y; acc[5][1] += wv.y * v2.x + wv.w * v2.y;
        acc[6][0] += wv.x * v2.z + wv.z * v2.w; acc[6][1] += wv.y * v2.z + wv.w * v2.w;
      }
    int rr = ly + 3, cc = lx + 3;
    size_t po = (size_t)y * W2 + x;
#pragma unroll
    for (int b = 0; b < 2; ++b) {
      float wspix = smem[OFF_WSL + (rr * SKW + cc) * 2 + b];
      float scs_sk = smem[OFF_RSKIP + b * SKPLANE + (y - cys) * TWs + (x - cxs)];
      float cs_sk = cs_skip[(size_t)(bb * 2 + b) * S2 + po];
      float xsk0 = x_skip[(size_t)(bb * 2 + b) * S2 + po];
      float xsk1 = x_skip[(size_t)((8 + bb) * 2 + b) * S2 + po];
      float scs_d = acc[0][b] + EPSF;
      float xup0 = acc[1][b] / scs_d, xup1 = acc[2][b] / scs_d;
      float cs_unf = acc[3][b], scs_unf = acc[4][b];
      float cs_f = acc[5][b] * wspix, scs_f = acc[6][b] * wspix;
      float s_unf = scs_unf / (cs_unf + EPSF);
      float om = 1.f - s_unf;
      float cs_up = scs_unf + om * cs_f;
      float scs_up = s_unf * scs_unf + om * scs_f;
      float s_pool = scs_up / (cs_up + EPSF);
      float wkd = smem[OFF_W + 192 + b], wks = smem[OFF_W + 194 + b];
      float wpd = (1.f - wkd) * s_pool;
      float invd = 1.f / (wkd + wpd + EPSF);
      out_x[(size_t)(bb * 2 + b) * S2 + po] = (wkd * xsk0 + wpd * xup0) * invd;
      out_x[(size_t)((8 + bb) * 2 + b) * S2 + po] = (wkd * xsk1 + wpd * xup1) * invd;
      float wps = (1.f - wks) * s_pool;
      float invs = 1.f / (wks + wps + EPSF);
      out_cs[(size_t)(bb * 2 + b) * S2 + po] = (wks * cs_sk + wps * cs_up) * invs;
      out_scs[(size_t)(bb * 2 + b) * S2 + po] = (wks * scs_sk + wps * scs_up) * invs;
    }
  }
}

extern "C" void kernel_launch(void* const* d_in, const int* in_sizes, int n_in,
                              void* d_out, int out_size, void* d_ws, size_t ws_size,
                              hipStream_t stream) {
  const float* x_pool = (const float*)d_in[0];
  const float* scs_pool = (const float*)d_in[1];
  const float* cs_pool = (const float*)d_in[2];
  const float* x_skip = (const float*)d_in[3];
  const float* scs_skip = (const float*)d_in[4];
  const float* cs_skip = (const float*)d_in[5];

  prep_kernel<<<1, 1, 0, stream>>>((const float*)d_in[6], (const float*)d_in[7],
                                   (const float*)d_in[8], (const float*)d_in[9],
                                   (const float*)d_in[10], (const float*)d_in[11],
                                   (float*)d_ws);

  float* out_x = (float*)d_out;
  float* out_scs = out_x + (size_t)16 * 2 * S2;
  float* out_cs = out_scs + (size_t)8 * 2 * S2;
  dim3 grid(WP / PW, HP / PH, 8);  // 19 x 22 x 8
  unpool_main<<<grid, 256, 0, stream>>>(x_pool, scs_pool, cs_pool, x_skip, scs_skip, cs_skip,
                                        (const float*)d_ws, out_x, out_scs, out_cs);
}